// GATv2_89996744720664
// MI455X (gfx1250) — compile-verified
//
#include <hip/hip_runtime.h>

#define HC 128            // H*C = 2*64
#define CPH 64            // channels per head
#define EDIM 32           // edge feature dim
#define NEG_SLOPE 0.2f
#define SA_STRIDE 130     // padded A-tile stride (kills LDS bank conflicts)

typedef __attribute__((ext_vector_type(2))) float v2f;
typedef __attribute__((ext_vector_type(8))) float v8f;

// ---------------------------------------------------------------------------
// WMMA GEMM: Y[M,128] = A[M,128] @ W[128,128], fp32, V_WMMA_F32_16X16X4_F32.
// One block = one 16-row tile x all 8 column tiles (8 waves). A tile + full W
// staged in LDS (8.3KB + 64KB of the 320KB/WGP pool).
// ---------------------------------------------------------------------------
__global__ void __launch_bounds__(256)
gemm128_wmma(const float* __restrict__ A, const float* __restrict__ W,
             float* __restrict__ Y, int M) {
  __shared__ float sW[HC * HC];          // 64 KB
  __shared__ float sA[16 * SA_STRIDE];   // padded 16x128 tile

  const int rowBase = blockIdx.x * 16;
  if (rowBase >= M) return;              // uniform across block

  // cooperative loads (coalesced float4)
  for (int i = threadIdx.x; i < HC * HC / 4; i += 256)
    ((float4*)sW)[i] = ((const float4*)W)[i];
  {
    const float* Ab = A + (size_t)rowBase * HC;
    for (int i = threadIdx.x; i < 16 * HC / 4; i += 256) {
      float4 v = ((const float4*)Ab)[i];
      int r = (i * 4) / HC, c = (i * 4) % HC;
      float* p = sA + r * SA_STRIDE + c;
      p[0] = v.x; p[1] = v.y; p[2] = v.z; p[3] = v.w;
    }
  }
  __syncthreads();

  const int lane = threadIdx.x & 31;
  const int wave = threadIdx.x >> 5;     // = colTile 0..7
  const int hi   = lane >> 4;            // half-wave select
  const int l15  = lane & 15;
  const int col  = wave * 16 + l15;

  v8f acc = {0.f, 0.f, 0.f, 0.f, 0.f, 0.f, 0.f, 0.f};
  #pragma unroll 8
  for (int kk = 0; kk < HC; kk += 4) {
    const int ka = kk + hi * 2;
    // A 16x4: lanes 0-15 -> K{ka,ka+1}=K{0,1}; lanes 16-31 -> K{2,3}
    v2f a = *(const v2f*)(sA + l15 * SA_STRIDE + ka);
    // B 4x16: VGPR0 rows {0,2}, VGPR1 rows {1,3} (striped across lanes)
    v2f b;
    b.x = sW[ka * HC + col];
    b.y = sW[(ka + 1) * HC + col];
    acc = __builtin_amdgcn_wmma_f32_16x16x4_f32(false, a, false, b,
                                                (short)0, acc, false, false);
  }
  // C/D layout: VGPR r -> M = r + hi*8, N = col
  #pragma unroll
  for (int r = 0; r < 8; ++r)
    Y[(size_t)(rowBase + hi * 8 + r) * HC + col] = acc[r];
}

// ---------------------------------------------------------------------------
// Per-layer init: m = -inf, denom = 0, out = bias broadcast.
// ---------------------------------------------------------------------------
__global__ void init_layer(float* __restrict__ m, float* __restrict__ denom,
                           float* __restrict__ out, const float* __restrict__ b,
                           int Nn) {
  int i = blockIdx.x * blockDim.x + threadIdx.x;
  if (i < Nn * 2) { m[i] = -__builtin_inff(); denom[i] = 0.f; }
  if (i < Nn * HC) out[i] = b[i & (HC - 1)];
}

// order-preserving float atomic max (guaranteed-available int atomics)
__device__ inline void atomicMaxF(float* addr, float val) {
  if (val >= 0.f)
    atomicMax((int*)addr, __float_as_int(val));
  else
    atomicMin((unsigned int*)addr, __float_as_uint(val));
}

// ---------------------------------------------------------------------------
// Edge logits: one wave per edge. Fuses ea = edge_attr @ We (LDS-cached We,
// shuffle-broadcast edge_attr), leaky-relu, att-dot, per-head half-wave
// reduction, logit store + segment atomic max.
// Lane owns channels ch0..ch0+3; lanes 0-15 = head 0, lanes 16-31 = head 1.
// ---------------------------------------------------------------------------
__global__ void __launch_bounds__(256)
edge_logits(const float* __restrict__ xl, const float* __restrict__ xr,
            const float* __restrict__ eattr, const float* __restrict__ We,
            const float* __restrict__ att, const int* __restrict__ src,
            const int* __restrict__ dst, float* __restrict__ logits,
            float* __restrict__ m, int Ee) {
  __shared__ float sWe[EDIM * HC];   // 16 KB
  __shared__ float sAtt[HC];
  for (int i = threadIdx.x; i < EDIM * HC; i += blockDim.x) sWe[i] = We[i];
  if (threadIdx.x < HC) sAtt[threadIdx.x] = att[threadIdx.x];
  __syncthreads();

  const int lane = threadIdx.x & 31;
  const int e = blockIdx.x * 8 + (threadIdx.x >> 5);
  if (e >= Ee) return;
  const int s = src[e], d = dst[e];

  // one coalesced load of this edge's 32 features, broadcast via shuffles
  const float ea_reg = eattr[(size_t)e * EDIM + lane];
  const int ch0 = lane * 4;
  float a0 = 0.f, a1 = 0.f, a2 = 0.f, a3 = 0.f;
  #pragma unroll
  for (int k = 0; k < EDIM; ++k) {
    const float w = __shfl(ea_reg, k, 32);
    const float* wp = sWe + k * HC + ch0;
    a0 += w * wp[0]; a1 += w * wp[1]; a2 += w * wp[2]; a3 += w * wp[3];
  }

  const float4 xj = *(const float4*)(xl + (size_t)s * HC + ch0);
  const float4 xi = *(const float4*)(xr + (size_t)d * HC + ch0);
  float t, sum = 0.f;
  t = xj.x + xi.x + a0; t = (t > 0.f) ? t : t * NEG_SLOPE; sum += t * sAtt[ch0 + 0];
  t = xj.y + xi.y + a1; t = (t > 0.f) ? t : t * NEG_SLOPE; sum += t * sAtt[ch0 + 1];
  t = xj.z + xi.z + a2; t = (t > 0.f) ? t : t * NEG_SLOPE; sum += t * sAtt[ch0 + 2];
  t = xj.w + xi.w + a3; t = (t > 0.f) ? t : t * NEG_SLOPE; sum += t * sAtt[ch0 + 3];

  // reduce within each 16-lane half (one head per half)
  #pragma unroll
  for (int off = 8; off; off >>= 1) sum += __shfl_xor(sum, off, 32);

  if ((lane & 15) == 0) {
    const int h = lane >> 4;
    logits[(size_t)e * 2 + h] = sum;
    atomicMaxF(m + (size_t)d * 2 + h, sum);
  }
}

// ---------------------------------------------------------------------------
// ex = exp(logit - m[dst]); denom[dst] += ex. One thread per (edge, head).
// ---------------------------------------------------------------------------
__global__ void exp_denom(const float* __restrict__ logits,
                          const float* __restrict__ m,
                          const int* __restrict__ dst,
                          float* __restrict__ exbuf,
                          float* __restrict__ denom, int Ee) {
  int i = blockIdx.x * blockDim.x + threadIdx.x;
  if (i >= Ee * 2) return;
  const int e = i >> 1, h = i & 1;
  const int d = dst[e];
  const float ex = __expf(logits[i] - m[(size_t)d * 2 + h]);
  exbuf[i] = ex;
  atomicAdd(denom + (size_t)d * 2 + h, ex);
}

// ---------------------------------------------------------------------------
// out[dst] += alpha * xl[src]. One wave per edge, 4 channels per lane.
// ---------------------------------------------------------------------------
__global__ void __launch_bounds__(256)
aggregate(const float* __restrict__ exbuf, const float* __restrict__ denom,
          const float* __restrict__ xl, const int* __restrict__ src,
          const int* __restrict__ dst, float* __restrict__ out, int Ee) {
  const int lane = threadIdx.x & 31;
  const int e = blockIdx.x * 8 + (threadIdx.x >> 5);
  if (e >= Ee) return;
  const int s = src[e], d = dst[e];
  const float a0 = exbuf[(size_t)e * 2 + 0] / (denom[(size_t)d * 2 + 0] + 1e-16f);
  const float a1 = exbuf[(size_t)e * 2 + 1] / (denom[(size_t)d * 2 + 1] + 1e-16f);
  const int ch0 = lane * 4;
  const float alpha = (ch0 < CPH) ? a0 : a1;     // 4 channels never straddle heads
  const float4 xj = *(const float4*)(xl + (size_t)s * HC + ch0);
  float* op = out + (size_t)d * HC + ch0;
  atomicAdd(op + 0, alpha * xj.x);
  atomicAdd(op + 1, alpha * xj.y);
  atomicAdd(op + 2, alpha * xj.z);
  atomicAdd(op + 3, alpha * xj.w);
}

__global__ void relu_copy(const float* __restrict__ in, float* __restrict__ out,
                          int n) {
  int i = blockIdx.x * blockDim.x + threadIdx.x;
  if (i < n) out[i] = fmaxf(in[i], 0.f);
}

// ---------------------------------------------------------------------------
extern "C" void kernel_launch(void* const* d_in, const int* in_sizes, int n_in,
                              void* d_out, int out_size, void* d_ws, size_t ws_size,
                              hipStream_t stream) {
  const float* x     = (const float*)d_in[0];
  const int*   ei    = (const int*)d_in[1];
  const float* eattr = (const float*)d_in[2];

  const int Nn = in_sizes[0] / HC;   // 50000
  const int Ee = in_sizes[1] / 2;    // 800000
  const int* src = ei;
  const int* dst = ei + Ee;

  // workspace layout (floats): ~116 MB total
  float* xl     = (float*)d_ws;
  float* xr     = xl     + (size_t)Nn * HC;
  float* hbuf   = xr     + (size_t)Nn * HC;
  float* accb   = hbuf   + (size_t)Nn * HC;
  float* logits = accb   + (size_t)Nn * HC;
  float* exbuf  = logits + (size_t)Ee * 2;
  float* mbuf   = exbuf  + (size_t)Ee * 2;
  float* denom  = mbuf   + (size_t)Nn * 2;

  const int rowTiles   = (Nn + 15) / 16;          // 3125 blocks
  const int initBlocks = (Nn * HC + 255) / 256;   // 25000
  const int edgeBlocks = (Ee + 7) / 8;            // 100000
  const int expBlocks  = (Ee * 2 + 255) / 256;    // 6250

  const float* cur = x;
  for (int l = 0; l < 3; ++l) {
    const float* Wl = (const float*)d_in[3 + l * 5 + 0];
    const float* Wr = (const float*)d_in[3 + l * 5 + 1];
    const float* We = (const float*)d_in[3 + l * 5 + 2];
    const float* at = (const float*)d_in[3 + l * 5 + 3];
    const float* b  = (const float*)d_in[3 + l * 5 + 4];
    float* target = (l < 2) ? accb : (float*)d_out;

    gemm128_wmma<<<rowTiles, 256, 0, stream>>>(cur, Wl, xl, Nn);
    gemm128_wmma<<<rowTiles, 256, 0, stream>>>(cur, Wr, xr, Nn);
    init_layer<<<initBlocks, 256, 0, stream>>>(mbuf, denom, target, b, Nn);
    edge_logits<<<edgeBlocks, 256, 0, stream>>>(xl, xr, eattr, We, at, src, dst,
                                                logits, mbuf, Ee);
    exp_denom<<<expBlocks, 256, 0, stream>>>(logits, mbuf, dst, exbuf, denom, Ee);
    aggregate<<<edgeBlocks, 256, 0, stream>>>(exbuf, denom, xl, src, dst, target, Ee);
    if (l < 2) {
      relu_copy<<<initBlocks, 256, 0, stream>>>(target, hbuf, Nn * HC);
      cur = hbuf;
    }
  }
}